// HybridQuantumModel_2525440770847
// MI455X (gfx1250) — compile-verified
//
#include <hip/hip_runtime.h>
#include <hip/hip_bf16.h>
#include <math.h>

typedef __attribute__((ext_vector_type(2))) float v2f;
typedef __attribute__((ext_vector_type(8))) float v8f;

#define PI_F 3.14159265358979323846f

// ---------------------------------------------------------------------------
// Init kernel: build the 16x16 real observable matrix M = Re(U^dag Z0 U)
// from q_params. One block of 256 threads; threads 0..15 each evolve one
// column of U (3 layers x [RX,RY,RZ per qubit + CNOT chain]), then all 256
// threads compute one entry of M each. Cost is negligible (launched once).
// ---------------------------------------------------------------------------
__global__ void qnn_init_M(const float* __restrict__ qp, float* __restrict__ Mg) {
    __shared__ float sUr[16][16];   // [row j][col t]
    __shared__ float sUi[16][16];
    const int t = threadIdx.x;
    if (t < 16) {
        float ur[16], ui[16];
        #pragma unroll
        for (int i = 0; i < 16; ++i) { ur[i] = (i == t) ? 1.f : 0.f; ui[i] = 0.f; }
        for (int l = 0; l < 3; ++l) {
            for (int q = 0; q < 4; ++q) {
                const float* p = qp + (l * 4 + q) * 3;
                const int m = 8 >> q;  // wire q <-> bit (3-q)
                // RX(p[0]): [[c,-is],[-is,c]]
                {
                    float c = cosf(0.5f * p[0]), s = sinf(0.5f * p[0]);
                    #pragma unroll
                    for (int i = 0; i < 16; ++i) if (!(i & m)) {
                        int ib = i | m;
                        float ar = ur[i], ai = ui[i], br = ur[ib], bi = ui[ib];
                        ur[i]  = c * ar + s * bi;  ui[i]  = c * ai - s * br;
                        ur[ib] = c * br + s * ai;  ui[ib] = c * bi - s * ar;
                    }
                }
                // RY(p[1]): [[c,-s],[s,c]]
                {
                    float c = cosf(0.5f * p[1]), s = sinf(0.5f * p[1]);
                    #pragma unroll
                    for (int i = 0; i < 16; ++i) if (!(i & m)) {
                        int ib = i | m;
                        float ar = ur[i], ai = ui[i], br = ur[ib], bi = ui[ib];
                        ur[i]  = c * ar - s * br;  ui[i]  = c * ai - s * bi;
                        ur[ib] = s * ar + c * br;  ui[ib] = s * ai + c * bi;
                    }
                }
                // RZ(p[2]): diag(e^{-i t/2}, e^{+i t/2})
                {
                    float c = cosf(0.5f * p[2]), s = sinf(0.5f * p[2]);
                    #pragma unroll
                    for (int i = 0; i < 16; ++i) if (!(i & m)) {
                        int ib = i | m;
                        float ar = ur[i], ai = ui[i], br = ur[ib], bi = ui[ib];
                        ur[i]  = c * ar + s * ai;  ui[i]  = c * ai - s * ar;
                        ur[ib] = c * br - s * bi;  ui[ib] = c * bi + s * br;
                    }
                }
            }
            // CNOT chain: control q, target q+1
            for (int q = 0; q < 3; ++q) {
                const int mc = 8 >> q, mt = 8 >> (q + 1);
                #pragma unroll
                for (int i = 0; i < 16; ++i) if ((i & mc) && !(i & mt)) {
                    int ib = i | mt;
                    float tr = ur[i]; ur[i] = ur[ib]; ur[ib] = tr;
                    float ti = ui[i]; ui[i] = ui[ib]; ui[ib] = ti;
                }
            }
        }
        #pragma unroll
        for (int i = 0; i < 16; ++i) { sUr[i][t] = ur[i]; sUi[i][t] = ui[i]; }
    }
    __syncthreads();
    // M[k][n] = sum_j z_j (Ur[j][k]Ur[j][n] + Ui[j][k]Ui[j][n]),  z_j = +-1
    const int k = threadIdx.x >> 4;
    const int n = threadIdx.x & 15;
    float acc = 0.f;
    #pragma unroll
    for (int j = 0; j < 16; ++j) {
        float zz = (j < 8) ? 1.f : -1.f;
        acc += zz * (sUr[j][k] * sUr[j][n] + sUi[j][k] * sUi[j][n]);
    }
    Mg[threadIdx.x] = acc;
}

// ---------------------------------------------------------------------------
// Main kernel: per lane build psi (real product state, 16 floats) in LDS,
// then per wave compute Y = P*M for two 16-batch tiles with
// V_WMMA_F32_16X16X4_F32 (K accumulated in 4 chunks), then z_b = <P_b, Y_b>
// via butterfly reduction, sigmoid, store. Memory-bound: 36B/element.
// ---------------------------------------------------------------------------
__global__ void __launch_bounds__(256)
qnn_main(const float* __restrict__ x, const float* __restrict__ Mg,
         float* __restrict__ out, int n) {
    __shared__ float sM[256];            // M, row-major 16x16
    __shared__ float sPsi[8][16][32];    // per wave: [component k][lane b]

    const int lane = threadIdx.x & 31;
    const int wave = threadIdx.x >> 5;
    const int waveBase = blockIdx.x * 256 + wave * 32;

    // cooperative load of M (one element per thread)
    sM[threadIdx.x] = Mg[threadIdx.x];

    // ---- per-lane encoding: psi = prod_i [cos t_i, sin t_i] ----
    int gid = waveBase + lane;
    int cg  = (gid < n) ? gid : (n - 1);
    const float4* xv = reinterpret_cast<const float4*>(x);
    float4 xa = xv[cg * 2 + 0];
    float4 xb = xv[cg * 2 + 1];
    float t0 = 0.5f * PI_F * (xa.x + xa.y);
    float t1 = 0.5f * PI_F * (xa.z + xa.w);
    float t2 = 0.5f * PI_F * (xb.x + xb.y);
    float t3 = 0.5f * PI_F * (xb.z + xb.w);
    float c0 = cosf(t0), s0 = sinf(t0);
    float c1 = cosf(t1), s1 = sinf(t1);
    float c2 = cosf(t2), s2 = sinf(t2);
    float c3 = cosf(t3), s3 = sinf(t3);
    // wire0 <-> bit 8, wire1 <-> bit 4, wire2 <-> bit 2, wire3 <-> bit 1
    #pragma unroll
    for (int i = 0; i < 16; ++i) {
        float v = ((i & 8) ? s0 : c0) * ((i & 4) ? s1 : c1) *
                  ((i & 2) ? s2 : c2) * ((i & 1) ? s3 : c3);
        sPsi[wave][i][lane] = v;
    }
    __syncthreads();

    // ---- WMMA: Y = P * M, two 16-row tiles per wave ----
    // A layout (32b 16x4): lane l holds A[M = l&15][K = v + 2*(l>>4)]
    // B layout (32b 4x16): lane l holds B[K = v + 2*(l>>4)][N = l&15]
    // D layout (32b 16x16): VGPR v, lane l -> Y[row v + 8*(l>>4)][col l&15]
    const int h = lane >> 4;
    const int j = lane & 15;
    const float* P = &sPsi[wave][0][0];   // stride 32 per component row

    v8f d0 = {}; v8f d1 = {};
    #pragma unroll
    for (int k = 0; k < 4; ++k) {
        const int K0 = 4 * k + 2 * h;
        const int K1 = K0 + 1;
        v2f a0, a1, b;
        a0.x = P[K0 * 32 + j];       a0.y = P[K1 * 32 + j];
        a1.x = P[K0 * 32 + 16 + j];  a1.y = P[K1 * 32 + 16 + j];
        b.x  = sM[K0 * 16 + j];      b.y  = sM[K1 * 16 + j];
        d0 = __builtin_amdgcn_wmma_f32_16x16x4_f32(false, a0, false, b,
                                                   (short)0, d0, false, false);
        d1 = __builtin_amdgcn_wmma_f32_16x16x4_f32(false, a1, false, b,
                                                   (short)0, d1, false, false);
    }

    // ---- z_b = sum_n P[b][n] * Y[b][n] : elementwise then 16-lane reduce ----
    float e0[8], e1[8];
    #pragma unroll
    for (int v = 0; v < 8; ++v) {
        const int row = v + 8 * h;
        e0[v] = P[row * 32 + j]      * d0[v];
        e1[v] = P[row * 32 + 16 + j] * d1[v];
    }
    #pragma unroll
    for (int m = 1; m < 16; m <<= 1) {
        #pragma unroll
        for (int v = 0; v < 8; ++v) {
            e0[v] += __shfl_xor(e0[v], m, 16);
            e1[v] += __shfl_xor(e1[v], m, 16);
        }
    }

    // lanes j<8 of each half write rows (j + 8h) of their tile
    if (j < 8) {
        const int row = j + 8 * h;
        int ob0 = waveBase + row;        // tile 0 (lanes 0..15's elements)
        int ob1 = waveBase + 16 + row;   // tile 1
        if (ob0 < n) {
            float z = e0[j];
            float a = 0.5f * (1.0f - z);
            out[ob0] = 1.0f / (1.0f + expf(-a));
        }
        if (ob1 < n) {
            float z = e1[j];
            float a = 0.5f * (1.0f - z);
            out[ob1] = 1.0f / (1.0f + expf(-a));
        }
    }
}

extern "C" void kernel_launch(void* const* d_in, const int* in_sizes, int n_in,
                              void* d_out, int out_size, void* d_ws, size_t ws_size,
                              hipStream_t stream) {
    const float* x  = (const float*)d_in[0];   // (B, 8) f32
    const float* qp = (const float*)d_in[1];   // (36,)  f32
    float* out = (float*)d_out;                // (B, 1) f32
    float* Mg  = (float*)d_ws;                 // 256 floats scratch
    const int n = out_size;

    hipLaunchKernelGGL(qnn_init_M, dim3(1), dim3(256), 0, stream, qp, Mg);
    const int blocks = (n + 255) / 256;
    hipLaunchKernelGGL(qnn_main, dim3(blocks), dim3(256), 0, stream, x, Mg, out, n);
}